// VoxelFeatureEncoder_52536039964808
// MI455X (gfx1250) — compile-verified
//
#include <hip/hip_runtime.h>

typedef __attribute__((ext_vector_type(2))) float v2f;
typedef __attribute__((ext_vector_type(8))) float v8f;

#define NB 2
#define NV 16000
#define NPOINT 32
#define NVOX (NB * NV)          // 32000 voxels
#define NPTS (NVOX * NPOINT)    // 1,024,000 points

// ---- workspace layout (float offsets) ----
#define WS_SUM1 0     // 16
#define WS_SQ1  16    // 16
#define WS_SUM2 32    // 32
#define WS_SQ2  64    // 32
#define WS_SUMF 96    // 64
#define WS_SQF  160   // 64   -> zero [0,224)
#define WS_S1   224
#define WS_B1   240
#define WS_S2   256
#define WS_B2   288
#define WS_SF   320
#define WS_BF   384   // total 448 floats

__device__ __forceinline__ void compute_feats(const float4& pt, const int4& cc, float f[8]) {
  // idx = coords[...,1:4][::-1] -> (c3,c2,c1); centers=(idx+0.5)*VOXEL+PC_MIN
  float cx = ((float)cc.w + 0.5f) * 0.16f + 0.0f;
  float cy = ((float)cc.z + 0.5f) * 0.16f - 39.68f;
  float cz = ((float)cc.y + 0.5f) * 4.0f - 3.0f;
  f[0] = pt.x; f[1] = pt.y; f[2] = pt.z;
  f[3] = pt.x - cx; f[4] = pt.y - cy; f[5] = pt.z - cz;
  f[6] = pt.w; f[7] = pt.z; // height == z
}

// ---------- pass 1: layer-1 pre-BN channel sums (scalar; tiny K) ----------
__global__ __launch_bounds__(256) void k_stats1(const float* __restrict__ vox,
                                                const int* __restrict__ coords,
                                                const float* __restrict__ W1,
                                                float* __restrict__ ws) {
  __shared__ float sacc[32];
  __shared__ float sW[16 * 8];
  for (int i = threadIdx.x; i < 32; i += blockDim.x) sacc[i] = 0.f;
  for (int i = threadIdx.x; i < 128; i += blockDim.x) sW[i] = W1[i];
  __syncthreads();

  float acc[16], acq[16];
  #pragma unroll
  for (int o = 0; o < 16; o++) { acc[o] = 0.f; acq[o] = 0.f; }

  int tid = blockIdx.x * blockDim.x + threadIdx.x;
  int stride = gridDim.x * blockDim.x;
  for (int p = tid; p < NPTS; p += stride) {
    int vxl = p >> 5;
    float4 pt = ((const float4*)vox)[p];
    int4 cc = ((const int4*)coords)[vxl];
    float f[8];
    compute_feats(pt, cc, f);
    #pragma unroll
    for (int o = 0; o < 16; o++) {
      float h = 0.f;
      #pragma unroll
      for (int c = 0; c < 8; c++) h = fmaf(sW[o * 8 + c], f[c], h);
      acc[o] += h;
      acq[o] = fmaf(h, h, acq[o]);
    }
  }
  #pragma unroll
  for (int o = 0; o < 16; o++) {
    atomicAdd(&sacc[o], acc[o]);
    atomicAdd(&sacc[16 + o], acq[o]);
  }
  __syncthreads();
  if (threadIdx.x < 32) atomicAdd(&ws[WS_SUM1 + threadIdx.x], sacc[threadIdx.x]);
}

// ---------- fold BN stats into per-channel scale/bias ----------
__global__ void k_bnprep(const float* __restrict__ sums, const float* __restrict__ sumsq,
                         const float* __restrict__ gamma, const float* __restrict__ beta,
                         float* __restrict__ scale, float* __restrict__ bias,
                         int C, float invN, float eps) {
  int o = threadIdx.x;
  if (o < C) {
    float m = sums[o] * invN;
    float var = sumsq[o] * invN - m * m;
    float s = gamma[o] * rsqrtf(var + eps);
    scale[o] = s;
    bias[o] = fmaf(-m, s, beta[o]);
  }
}

// ---------- fused VFE passes (WMMA). FINAL=false: BN2 stats. FINAL=true: output + final stats ----------
template <bool FINAL>
__global__ __launch_bounds__(256) void k_pass(const float* __restrict__ vox,
                                              const int* __restrict__ coords,
                                              const int* __restrict__ nump,
                                              const float* __restrict__ W1,
                                              const float* __restrict__ W2,
                                              const float* __restrict__ Wf,
                                              float* __restrict__ ws,
                                              float* __restrict__ out) {
  __shared__ float lds_f[8 * 32 * 8];    // per-wave feats [point][8]
  __shared__ float lds_x2[8 * 32 * 32];  // per-wave layer-2 input [point][32]
  __shared__ float lds_vf[16 * 64];      // per-chunk voxel features (rows 8..15 stay zero)
  __shared__ float sstat[128];

  const int lane = threadIdx.x & 31;
  const int wave = threadIdx.x >> 5;
  const int n = lane & 15;  // N column / A-row index
  const int hi = lane >> 4; // half-wave

  for (int i = threadIdx.x; i < 128; i += blockDim.x) sstat[i] = 0.f;
  for (int i = threadIdx.x; i < 16 * 64; i += blockDim.x) lds_vf[i] = 0.f;
  __syncthreads();

  float* myf = lds_f + wave * (32 * 8);
  float* myx = lds_x2 + wave * (32 * 32);

  // ---- preload B fragments (weights): B[k][n] = W[n][k]; lane holds K = {2*hi, 2*hi+1} of each 4-step
  v2f bw1[2], bw2[2][8], bwf[16];
  #pragma unroll
  for (int kb = 0; kb < 2; kb++) bw1[kb] = *(const v2f*)(W1 + n * 8 + 4 * kb + 2 * hi);
  #pragma unroll
  for (int nt = 0; nt < 2; nt++)
    #pragma unroll
    for (int kb = 0; kb < 8; kb++)
      bw2[nt][kb] = *(const v2f*)(W2 + (nt * 16 + n) * 32 + 4 * kb + 2 * hi);
  if (FINAL) {
    int fnt = wave & 3;
    #pragma unroll
    for (int kb = 0; kb < 16; kb++)
      bwf[kb] = *(const v2f*)(Wf + (fnt * 16 + n) * 64 + 4 * kb + 2 * hi);
  }

  const float s1v = ws[WS_S1 + n], b1v = ws[WS_B1 + n];
  float s2v[2] = {0.f, 0.f}, b2v[2] = {0.f, 0.f};
  if (FINAL) {
    s2v[0] = ws[WS_S2 + n];      s2v[1] = ws[WS_S2 + 16 + n];
    b2v[0] = ws[WS_B2 + n];      b2v[1] = ws[WS_B2 + 16 + n];
  }

  float accS[2] = {0.f, 0.f}, accQ[2] = {0.f, 0.f}; // BN2 stats partials
  float accSF = 0.f, accQF = 0.f;                   // final BN stats partials

  const int nchunks = NVOX / 8; // 8 voxels (1/wave) per chunk
  for (int chunk = blockIdx.x; chunk < nchunks; chunk += gridDim.x) {
    const int vxl = chunk * 8 + wave;
    float4 pt = ((const float4*)vox)[vxl * NPOINT + lane];
    int4 cc = ((const int4*)coords)[vxl];
    int np_ = nump[vxl];

    float f[8];
    compute_feats(pt, cc, f);
    *(float4*)(myf + lane * 8 + 0) = make_float4(f[0], f[1], f[2], f[3]);
    *(float4*)(myf + lane * 8 + 4) = make_float4(f[4], f[5], f[6], f[7]);
    __syncthreads();

    // ---- layer 1: (32x8) @ (8x16) via 16x16x4 f32 WMMA
    v8f h1t[2];
    #pragma unroll
    for (int mt = 0; mt < 2; mt++) {
      v8f acc = {};
      #pragma unroll
      for (int kb = 0; kb < 2; kb++) {
        v2f a = *(const v2f*)(myf + (mt * 16 + n) * 8 + 4 * kb + 2 * hi);
        acc = __builtin_amdgcn_wmma_f32_16x16x4_f32(false, a, false, bw1[kb], (short)0, acc,
                                                    false, false);
      }
      h1t[mt] = acc;
    }

    // ---- BN1 + relu + voxel max + concat + mask -> x2 in LDS
    float hval[2][8];
    float mxp = 0.f; // relu output >= 0
    #pragma unroll
    for (int mt = 0; mt < 2; mt++)
      #pragma unroll
      for (int r = 0; r < 8; r++) {
        float h = fmaxf(fmaf(s1v, h1t[mt][r], b1v), 0.f);
        hval[mt][r] = h;
        mxp = fmaxf(mxp, h);
      }
    mxp = fmaxf(mxp, __shfl_xor(mxp, 16));
    #pragma unroll
    for (int mt = 0; mt < 2; mt++)
      #pragma unroll
      for (int r = 0; r < 8; r++) {
        int p = mt * 16 + 8 * hi + r; // D layout: M = r + 8*hi
        float m = (p < np_) ? 1.f : 0.f;
        myx[p * 32 + n] = hval[mt][r] * m;
        myx[p * 32 + 16 + n] = mxp * m;
      }
    __syncthreads();

    // ---- layer 2: (32x32) @ (32x32)
    v8f h2t[2][2];
    #pragma unroll
    for (int mt = 0; mt < 2; mt++) {
      v2f a[8];
      #pragma unroll
      for (int kb = 0; kb < 8; kb++)
        a[kb] = *(const v2f*)(myx + (mt * 16 + n) * 32 + 4 * kb + 2 * hi);
      #pragma unroll
      for (int nt = 0; nt < 2; nt++) {
        v8f acc = {};
        #pragma unroll
        for (int kb = 0; kb < 8; kb++)
          acc = __builtin_amdgcn_wmma_f32_16x16x4_f32(false, a[kb], false, bw2[nt][kb], (short)0,
                                                      acc, false, false);
        h2t[mt][nt] = acc;
      }
    }

    if (!FINAL) {
      // ---- accumulate BN2 sums (pre-BN h2, masked rows already zero)
      #pragma unroll
      for (int nt = 0; nt < 2; nt++) {
        float s = 0.f, q = 0.f;
        #pragma unroll
        for (int mt = 0; mt < 2; mt++)
          #pragma unroll
          for (int r = 0; r < 8; r++) {
            float h = h2t[mt][nt][r];
            s += h;
            q = fmaf(h, h, q);
          }
        accS[nt] += s;
        accQ[nt] += q;
      }
    } else {
      // ---- BN2 + relu + voxel max/concat/mask -> vfeat[64] per voxel
      float mx2[2] = {0.f, 0.f};
      float vfh[2] = {0.f, 0.f};
      #pragma unroll
      for (int nt = 0; nt < 2; nt++) {
        #pragma unroll
        for (int mt = 0; mt < 2; mt++)
          #pragma unroll
          for (int r = 0; r < 8; r++) {
            float h = fmaxf(fmaf(s2v[nt], h2t[mt][nt][r], b2v[nt]), 0.f);
            mx2[nt] = fmaxf(mx2[nt], h);
            int p = mt * 16 + 8 * hi + r;
            if (p < np_) vfh[nt] = fmaxf(vfh[nt], h);
          }
        mx2[nt] = fmaxf(mx2[nt], __shfl_xor(mx2[nt], 16));
        vfh[nt] = fmaxf(vfh[nt], __shfl_xor(vfh[nt], 16));
      }
      float anyv = (np_ > 0) ? 1.f : 0.f;
      if (hi == 0) {
        lds_vf[wave * 64 + n]      = vfh[0];
        lds_vf[wave * 64 + 16 + n] = vfh[1];
        lds_vf[wave * 64 + 32 + n] = mx2[0] * anyv;
        lds_vf[wave * 64 + 48 + n] = mx2[1] * anyv;
      }
      __syncthreads();

      // ---- final GEMM: (8x64) @ (64x64); waves 0..3 each own one N-tile
      if (wave < 4) {
        v8f acc = {};
        #pragma unroll
        for (int kb = 0; kb < 16; kb++) {
          v2f a = *(const v2f*)(lds_vf + n * 64 + 4 * kb + 2 * hi);
          acc = __builtin_amdgcn_wmma_f32_16x16x4_f32(false, a, false, bwf[kb], (short)0, acc,
                                                      false, false);
        }
        if (hi == 0) { // valid output rows are M = 0..7
          #pragma unroll
          for (int r = 0; r < 8; r++) {
            float g = acc[r];
            out[(chunk * 8 + r) * 64 + wave * 16 + n] = g;
            accSF += g;
            accQF = fmaf(g, g, accQF);
          }
        }
      }
      __syncthreads();
    }
  }

  // ---- flush stats: registers -> LDS atomics -> global atomics
  __syncthreads();
  if (!FINAL) {
    atomicAdd(&sstat[n], accS[0]);
    atomicAdd(&sstat[16 + n], accS[1]);
    atomicAdd(&sstat[32 + n], accQ[0]);
    atomicAdd(&sstat[48 + n], accQ[1]);
    __syncthreads();
    for (int i = threadIdx.x; i < 64; i += blockDim.x) atomicAdd(&ws[WS_SUM2 + i], sstat[i]);
  } else {
    if (wave < 4 && hi == 0) {
      atomicAdd(&sstat[wave * 16 + n], accSF);
      atomicAdd(&sstat[64 + wave * 16 + n], accQF);
    }
    __syncthreads();
    for (int i = threadIdx.x; i < 128; i += blockDim.x) atomicAdd(&ws[WS_SUMF + i], sstat[i]);
  }
}

// ---------- final elementwise batchnorm ----------
__global__ __launch_bounds__(256) void k_norm(float* __restrict__ out,
                                              const float* __restrict__ scale,
                                              const float* __restrict__ bias, int nElem) {
  int i = blockIdx.x * blockDim.x + threadIdx.x;
  if (i < nElem) {
    int c = i & 63;
    out[i] = fmaf(scale[c], out[i], bias[c]);
  }
}

extern "C" void kernel_launch(void* const* d_in, const int* in_sizes, int n_in,
                              void* d_out, int out_size, void* d_ws, size_t ws_size,
                              hipStream_t stream) {
  (void)in_sizes; (void)n_in; (void)out_size; (void)ws_size;
  const float* vox    = (const float*)d_in[0];
  const int*   coords = (const int*)d_in[1];
  const int*   nump   = (const int*)d_in[2];
  const float* W1     = (const float*)d_in[3];
  const float* g1     = (const float*)d_in[4];
  const float* b1     = (const float*)d_in[5];
  const float* W2     = (const float*)d_in[6];
  const float* g2     = (const float*)d_in[7];
  const float* b2     = (const float*)d_in[8];
  const float* Wf     = (const float*)d_in[9];
  const float* gf     = (const float*)d_in[10];
  const float* bf     = (const float*)d_in[11];
  float* out = (float*)d_out;
  float* ws  = (float*)d_ws;

  hipMemsetAsync(ws, 0, 224 * sizeof(float), stream);
  k_stats1<<<1024, 256, 0, stream>>>(vox, coords, W1, ws);
  k_bnprep<<<1, 64, 0, stream>>>(ws + WS_SUM1, ws + WS_SQ1, g1, b1, ws + WS_S1, ws + WS_B1,
                                 16, 1.0f / (float)NPTS, 1e-3f);
  k_pass<false><<<250, 256, 0, stream>>>(vox, coords, nump, W1, W2, Wf, ws, out);
  k_bnprep<<<1, 64, 0, stream>>>(ws + WS_SUM2, ws + WS_SQ2, g2, b2, ws + WS_S2, ws + WS_B2,
                                 32, 1.0f / (float)NPTS, 1e-3f);
  k_pass<true><<<250, 256, 0, stream>>>(vox, coords, nump, W1, W2, Wf, ws, out);
  k_bnprep<<<1, 64, 0, stream>>>(ws + WS_SUMF, ws + WS_SQF, gf, bf, ws + WS_SF, ws + WS_BF,
                                 64, 1.0f / (float)NVOX, 1e-5f);
  k_norm<<<(NVOX * 64 + 255) / 256, 256, 0, stream>>>(out, ws + WS_SF, ws + WS_BF, NVOX * 64);
}